// ModelWithScan_91130616086856
// MI455X (gfx1250) — compile-verified
//
#include <hip/hip_runtime.h>
#include <hip/hip_bf16.h>
#include <stdint.h>

// Problem constants (from reference): R=1024, C=131072, NSEG=1024, CROP=48
#define R_ROWS 1024
#define C_COLS 131072
#define NSEG_  1024

// ---------------------------------------------------------------------------
// Kernel 1: build the flat column-index map.
//   lens[s]  = min(splits[s+1]-splits[s], crop)
//   offs     = exclusive scan of lens
//   idxmap[offs[s]+k] = splits[s]+k   for k < lens[s]
// One block of 1024 threads (one thread per segment); Hillis-Steele scan in LDS.
// ---------------------------------------------------------------------------
__global__ void build_idx_kernel(const int* __restrict__ splits,
                                 const int* __restrict__ crop_p,
                                 int* __restrict__ idxmap) {
    __shared__ int scan[NSEG_];
    const int t = threadIdx.x;            // 0..1023
    const int crop = *crop_p;

    const int a = splits[t];
    const int b = splits[t + 1];
    int len = b - a;
    if (len > crop) len = crop;
    if (len < 0)    len = 0;

    scan[t] = len;
    __syncthreads();

    // Inclusive Hillis-Steele scan over 1024 elements.
    #pragma unroll
    for (int d = 1; d < NSEG_; d <<= 1) {
        int add = (t >= d) ? scan[t - d] : 0;
        __syncthreads();
        scan[t] += add;
        __syncthreads();
    }

    const int off = scan[t] - len;        // exclusive prefix
    for (int k = 0; k < len; ++k) {
        idxmap[off + k] = a + k;
    }
}

// ---------------------------------------------------------------------------
// Kernel 2: bandwidth-bound gather via the CDNA5 async global<->LDS pipeline.
// Grid: (ceil(total/COLS), R).  Block handles one row x COLS columns.
// Phase 1: per-lane gather loads HBM -> LDS  (global_load_async_to_lds_b32,
//          non-temporal: each W element is consumed exactly once)
// Phase 2: contiguous stores LDS -> HBM      (global_store_async_from_lds_b32,
//          non-temporal: out is write-once streaming)
// idxmap loads stay regular-temporal so the 190 KB map survives in L2 across
// all 1024 row passes.  Synchronized per-wave with s_wait_asynccnt; each
// thread owns its LDS slots, so no block barrier is needed.
// ---------------------------------------------------------------------------
constexpr int BLK = 256;      // 8 waves of 32
constexpr int U   = 16;       // elements per thread -> 16 outstanding async
                              // loads per wave (ASYNCcnt is 6 bits, max 63)
constexpr int COLS = BLK * U; // 4096 columns per block -> 16 KB LDS

__global__ void gather_async_kernel(const float* __restrict__ W,
                                    const int* __restrict__ idxmap,
                                    float* __restrict__ out,
                                    int total) {
    __shared__ float lbuf[COLS];

    const int r  = blockIdx.y;
    const int c0 = blockIdx.x * COLS;
    const int t  = threadIdx.x;

    const float* Wrow = W   + (size_t)r * C_COLS;
    float*       Orow = out + (size_t)r * total;

    // Generic pointer to LDS: low 32 bits carry the LDS byte offset
    // (LDS aperture occupies the high word of the flat address).
    const unsigned lds0 = (unsigned)(uintptr_t)(void*)lbuf;

    // ---- Phase 1: async gather loads into this thread's LDS slots ----
    #pragma unroll
    for (int u = 0; u < U; ++u) {
        const int j = c0 + u * BLK + t;
        if (j < total) {
            const int col = idxmap[j];
            const unsigned long long ga =
                (unsigned long long)(uintptr_t)(Wrow + col);
            const unsigned lo = lds0 + (unsigned)(u * BLK + t) * 4u;
            asm volatile("global_load_async_to_lds_b32 %0, %1, off th:TH_LOAD_NT"
                         :: "v"(lo), "v"(ga) : "memory");
        }
    }

    // Drain loads before the dependent LDS reads of the async stores.
    asm volatile("s_wait_asynccnt 0x0" ::: "memory");

    // ---- Phase 2: async contiguous non-temporal stores LDS -> out ----
    #pragma unroll
    for (int u = 0; u < U; ++u) {
        const int j = c0 + u * BLK + t;
        if (j < total) {
            const unsigned long long ga =
                (unsigned long long)(uintptr_t)(Orow + j);
            const unsigned lo = lds0 + (unsigned)(u * BLK + t) * 4u;
            asm volatile("global_store_async_from_lds_b32 %0, %1, off th:TH_STORE_NT"
                         :: "v"(ga), "v"(lo) : "memory");
        }
    }

    // Ensure stores are complete before the wave retires (S_ENDPGM also
    // waits, but be explicit).
    asm volatile("s_wait_asynccnt 0x0" ::: "memory");
}

// ---------------------------------------------------------------------------
// kernel_launch
//   d_in[0] = W       (float32, R*C)
//   d_in[1] = splits  (int32, NSEG+1)
//   d_in[2] = crop    (int32 scalar)
//   d_out   = out     (float32, R * total), total = out_size / R
//   d_ws    = idxmap  (int32, total)  — ~190 KB scratch
// ---------------------------------------------------------------------------
extern "C" void kernel_launch(void* const* d_in, const int* in_sizes, int n_in,
                              void* d_out, int out_size, void* d_ws, size_t ws_size,
                              hipStream_t stream) {
    const float* W      = (const float*)d_in[0];
    const int*   splits = (const int*)d_in[1];
    const int*   crop   = (const int*)d_in[2];
    float*       out    = (float*)d_out;
    int*         idxmap = (int*)d_ws;

    const int total = out_size / R_ROWS;

    build_idx_kernel<<<1, NSEG_, 0, stream>>>(splits, crop, idxmap);

    if (total > 0) {
        dim3 grid((total + COLS - 1) / COLS, R_ROWS);
        gather_async_kernel<<<grid, BLK, 0, stream>>>(W, idxmap, out, total);
    }
}